// LocalAggregator_34144990003658
// MI455X (gfx1250) — compile-verified
//
#include <hip/hip_runtime.h>

typedef __attribute__((ext_vector_type(2))) float v2f;
typedef __attribute__((ext_vector_type(4))) float v4f;
typedef __attribute__((ext_vector_type(8))) float v8f;

#define LEAKY  0.2f
#define MASKV  -9.0e15f
#define HS     132   // LDS row stride for hidden tile (128 + 4 pad -> conflict-free frag reads)
#define AS     68    // LDS row stride for alpha tile  (64 + 4 pad)

__launch_bounds__(128)
__global__ void gat_local_agg_kernel(const float* __restrict__ hidden,
                                     const int*   __restrict__ adj,
                                     const float* __restrict__ a,
                                     float*       __restrict__ out) {
  __shared__ __align__(16) float sh_h[64 * HS];      // hidden[b], padded rows
  __shared__ __align__(16) float sh_a[4 * 128];      // a_0..a_3
  __shared__ __align__(16) float sh_alpha[64 * AS];  // logits -> exp()
  __shared__ float sh_rinv[64];                      // 1/row_sum

  const int b    = blockIdx.x;
  const int tid  = threadIdx.x;
  const int w    = tid >> 5;    // wave id 0..3 == i-tile
  const int lane = tid & 31;
  const int half = lane >> 4;   // lane group 0/1
  const int lx   = lane & 15;

  const float* hb = hidden + (size_t)b * (64 * 128);

  // ---- stage hidden[b] (float4) into padded LDS ----
  for (int p = tid; p < 64 * 32; p += 128) {
    int row = p >> 5, c4 = p & 31;
    v4f v = *((const v4f*)hb + p);
    *(v4f*)&sh_h[row * HS + c4 * 4] = v;
  }
  // ---- stage a (4x128 floats == 128 float4, one per thread) ----
  {
    v4f v = *((const v4f*)a + tid);
    *(v4f*)&sh_a[tid * 4] = v;
  }
  __syncthreads();

  const int i0 = w * 16 + lx;   // A-fragment row (M) for this lane

  // ---- Phase A: e_k = (H .* a_k) @ H^T for i-tile w, all 4 j-tiles ----
  const int* adjb = adj + (size_t)b * 4096;
  for (int jt = 0; jt < 4; ++jt) {
    const int j0 = jt * 16 + lx;          // B-fragment column (N)
    v8f acc0 = {}, acc1 = {}, acc2 = {}, acc3 = {};
    for (int kb = 0; kb < 128; kb += 4) {
      int kk = kb + 2 * half;             // K index for VGPR0 of this lane half
      float h0 = sh_h[i0 * HS + kk];
      float h1 = sh_h[i0 * HS + kk + 1];
      v2f Bf = { sh_h[j0 * HS + kk], sh_h[j0 * HS + kk + 1] };
      v2f A0 = { h0 * sh_a[0 * 128 + kk], h1 * sh_a[0 * 128 + kk + 1] };
      v2f A1 = { h0 * sh_a[1 * 128 + kk], h1 * sh_a[1 * 128 + kk + 1] };
      v2f A2 = { h0 * sh_a[2 * 128 + kk], h1 * sh_a[2 * 128 + kk + 1] };
      v2f A3 = { h0 * sh_a[3 * 128 + kk], h1 * sh_a[3 * 128 + kk + 1] };
      acc0 = __builtin_amdgcn_wmma_f32_16x16x4_f32(false, A0, false, Bf, (short)0, acc0, false, false);
      acc1 = __builtin_amdgcn_wmma_f32_16x16x4_f32(false, A1, false, Bf, (short)0, acc1, false, false);
      acc2 = __builtin_amdgcn_wmma_f32_16x16x4_f32(false, A2, false, Bf, (short)0, acc2, false, false);
      acc3 = __builtin_amdgcn_wmma_f32_16x16x4_f32(false, A3, false, Bf, (short)0, acc3, false, false);
    }
    // LeakyReLU + adj selection -> masked logits into LDS
    // C/D layout: VGPR r -> (M = r + 8*half, N = lx)
    #pragma unroll
    for (int r = 0; r < 8; ++r) {
      int irow = w * 16 + r + 8 * half;
      int jc   = jt * 16 + lx;
      int av   = adjb[irow * 64 + jc];
      float e0 = acc0[r]; e0 = e0 >= 0.f ? e0 : LEAKY * e0;
      float e1 = acc1[r]; e1 = e1 >= 0.f ? e1 : LEAKY * e1;
      float e2 = acc2[r]; e2 = e2 >= 0.f ? e2 : LEAKY * e2;
      float e3 = acc3[r]; e3 = e3 >= 0.f ? e3 : LEAKY * e3;
      float val = MASKV;
      val = (av == 1) ? e0 : val;
      val = (av == 2) ? e1 : val;
      val = (av == 3) ? e2 : val;
      val = (av == 4) ? e3 : val;
      sh_alpha[irow * AS + jc] = val;
    }
  }
  __syncthreads();

  // ---- row softmax: store exp(x - max), keep reciprocal row sum ----
  if (tid < 64) {
    float mx = -3.0e38f;
    for (int j2 = 0; j2 < 64; ++j2) mx = fmaxf(mx, sh_alpha[tid * AS + j2]);
    float s = 0.f;
    for (int j2 = 0; j2 < 64; ++j2) {
      float ev = __expf(sh_alpha[tid * AS + j2] - mx);
      sh_alpha[tid * AS + j2] = ev;
      s += ev;
    }
    sh_rinv[tid] = 1.0f / s;
  }
  __syncthreads();

  // ---- Phase B: out[b] = softmax(alpha) @ H ----
  float* ob = out + (size_t)b * (64 * 128);
  for (int nt = 0; nt < 8; ++nt) {
    const int d0 = nt * 16 + lx;          // B-fragment column (N = feature d)
    v8f acc = {};
    for (int kb = 0; kb < 64; kb += 4) {
      int kk = kb + 2 * half;             // K index (contraction over j)
      v2f Af = { sh_alpha[i0 * AS + kk], sh_alpha[i0 * AS + kk + 1] };
      v2f Bf = { sh_h[kk * HS + d0],     sh_h[(kk + 1) * HS + d0] };
      acc = __builtin_amdgcn_wmma_f32_16x16x4_f32(false, Af, false, Bf, (short)0, acc, false, false);
    }
    #pragma unroll
    for (int r = 0; r < 8; ++r) {
      int irow = w * 16 + r + 8 * half;
      ob[irow * 128 + d0] = acc[r] * sh_rinv[irow];
    }
  }
}

extern "C" void kernel_launch(void* const* d_in, const int* in_sizes, int n_in,
                              void* d_out, int out_size, void* d_ws, size_t ws_size,
                              hipStream_t stream) {
  const float* hidden = (const float*)d_in[0];   // [B,64,128] f32
  const int*   adj    = (const int*)d_in[1];     // [B,64,64]  i32
  const float* a      = (const float*)d_in[2];   // [4,128]    f32
  float* out = (float*)d_out;                    // [B,64,128] f32
  const int B = in_sizes[0] / (64 * 128);        // 512
  gat_local_agg_kernel<<<B, 128, 0, stream>>>(hidden, adj, a, out);
}